// MCPBRNN_GWVariant_Routing_10196252360847
// MI455X (gfx1250) — compile-verified
//
#include <hip/hip_runtime.h>
#include <stdint.h>

// MCPBRNN scan: c' = (1 - oo(c))*c + u, oo(c) = oo1*sigmoid(A*c + Bc).
// Nonlinear in c -> strictly sequential over B*S = 4M steps. Roofline: 16MB of
// x is 0.7us at 23.3 TB/s; the dependent 5-op VALU chain (fma->exp->add->rcp->
// fma) dominates at ~4M * ~20cyc. So: one wave32 scans out of LDS (ds_load
// latency, fixed) while CDNA5 async engines double-buffer the next row:
//   even rows: TENSOR_LOAD_TO_LDS (TDM descriptor, TENSORcnt)
//   odd  rows: GLOBAL_LOAD_ASYNC_TO_LDS_B128 (ASYNCcnt)
// Alternating engines means each "wait cnt 0" only drains the current row's
// engine while the other engine is still prefetching -> full overlap, no
// barriers (a single wave both issues and consumes).

#define MAX_S 2048  // LDS double buffer: 2 x MAX_S floats (16KB)

typedef uint32_t u32x4 __attribute__((ext_vector_type(4)));
typedef uint32_t u32x8 __attribute__((ext_vector_type(8)));

// One scan step; off-chain ops (oo1n*c, c+u) overlap the transcendentals.
#define ROW_STEP(u_expr)                                                     \
    {                                                                        \
        const float u_  = (u_expr);                                         \
        const float e_  = __builtin_amdgcn_exp2f(__builtin_fmaf(c, A2, B2));\
        const float rc_ = __builtin_amdgcn_rcpf(1.0f + e_);                 \
        c = __builtin_fmaf(oo1n * c, rc_, c + u_);                          \
    }

// Last step of a row: capture outputs from the PRE-update carry, then update.
#define ROW_LAST(u_expr)                                                     \
    {                                                                        \
        const float u_  = (u_expr);                                         \
        const float e_  = __builtin_amdgcn_exp2f(__builtin_fmaf(c, A2, B2));\
        const float rc_ = __builtin_amdgcn_rcpf(1.0f + e_);                 \
        if (lane == 0) {                                                    \
            const float oo_ = oo1 * rc_;                                    \
            h_out [tl + r] = oo_ * c;                                       \
            c_out [tl + r] = c;                                             \
            oo_out[tl + r] = oo_;                                           \
            f_out [tl + r] = 1.0f - oo_;                                    \
        }                                                                   \
        c = __builtin_fmaf(oo1n * c, rc_, c + u_);                          \
    }

__launch_bounds__(32)
__global__ void mcpbrnn_scan_kernel(const float* __restrict__ x,
                                    const float* __restrict__ c_mean,
                                    const float* __restrict__ c_std,
                                    const float* __restrict__ ini_c,
                                    const float* __restrict__ w_yom,
                                    const float* __restrict__ w_yfm,
                                    const float* __restrict__ b0,
                                    const float* __restrict__ wb1,
                                    const int*   __restrict__ time_lag_p,
                                    float*       __restrict__ out,
                                    int B, int S)
{
    __shared__ float lds[2 * MAX_S];
    const int lane = threadIdx.x;  // wave32

    // Zero all 4*B outputs (harness poisons d_out; rows < time_lag must be 0).
    for (int i = lane; i < 4 * B; i += 32) out[i] = 0.0f;

    const int tl    = time_lag_p[0];
    const int nrows = B - tl;
    if (nrows <= 0) return;

    // Hoisted scalars. oo = oo1 / (1 + exp(-(A*c + Bc))); exp(-t) = exp2(A2*c+B2).
    const float so   = c_std[0];
    const float mo   = c_mean[0];
    const float w1   = wb1[0];
    const float bias = b0[0];
    const float oo1  = 1.0f / (1.0f + __expf(w_yfm[0] - w_yom[0]));  // sigmoid(wy-wf)
    const float oo1n = -oo1;
    const float A    = w1 / so;
    const float Bc   = bias - mo * A;
    const float L2E  = 1.4426950408889634f;
    const float A2   = -A * L2E;
    const float B2   = -Bc * L2E;
    float c = ini_c[0];

    float* h_out  = out;
    float* c_out  = out + B;
    float* oo_out = out + 2 * B;
    float* f_out  = out + 3 * B;

    // Low 32 bits of a flat __shared__ address are the LDS byte offset
    // (ISA 10.2) -- what VDST / D#.lds_addr want.
    const uint32_t lds_base  = (uint32_t)(uintptr_t)&lds[0];
    const uint32_t row_bytes = (uint32_t)S * 4u;

    // --- TDM descriptor group 1 (constant across rows), ISA 8.4 layout ---
    // data_size=2 (4B), tensor = S x 1 f32, tile = S x 1, stride0 = S.
    u32x8 g1;
    g1[0] = 0x00020000u;                              // wg_mask=0, data_size=4B
    g1[1] = ((uint32_t)S & 0xFFFFu) << 16;            // tensor_dim0[15:0]
    g1[2] = (((uint32_t)S >> 16) & 0xFFFFu) | (1u << 16); // dim0[31:16], tensor_dim1=1
    g1[3] = ((uint32_t)S & 0xFFFFu) << 16;            // tensor_dim1 hi=0, tile_dim0=S
    g1[4] = 1u;                                       // tile_dim1=1, tile_dim2=0
    g1[5] = (uint32_t)S;                              // tensor_dim0_stride[31:0]
    g1[6] = 0u;                                       // stride0 hi, stride1 lo
    g1[7] = 0u;                                       // stride1 hi

    // Issue one row via the Tensor Data Mover (1 instruction, TENSORcnt).
    auto issue_tdm = [&](int r, int buf) {
        const uint64_t ga = (uint64_t)(uintptr_t)(x + (size_t)(tl + r) * S);
        u32x4 g0;
        g0[0] = 1u;                                   // count=1, user descriptor
        g0[1] = lds_base + (uint32_t)buf * row_bytes; // D#.lds_addr
        g0[2] = (uint32_t)ga;                         // global_addr[31:0]
        g0[3] = ((uint32_t)(ga >> 32) & 0x01FFFFFFu)  // global_addr[56:32]
              | 0x80000000u;                          // type=2 ("image")
        asm volatile("tensor_load_to_lds %0, %1" :: "s"(g0), "s"(g1) : "memory");
    };

    // Issue one row via per-lane async B128 copies (S/128 instrs, ASYNCcnt).
    auto issue_async = [&](int r, int buf) {
        const char* ga = (const char*)(x + (size_t)(tl + r) * S) + (size_t)lane * 16u;
        uint32_t    lo = lds_base + (uint32_t)buf * row_bytes + (uint32_t)lane * 16u;
        const int   n  = S >> 7;  // B128 ops per lane (needs S % 128 == 0)
        for (int k = 0; k < n; ++k) {
            asm volatile("global_load_async_to_lds_b128 %0, %1, off"
                         :: "v"(lo), "v"(ga) : "memory");
            lo += 512u;  // 32 lanes * 16B
            ga += 512;
        }
    };

    const bool pipelined = ((S & 127) == 0) && (S <= MAX_S);
    if (pipelined) {
        issue_tdm(0, 0);  // prologue: row 0 on the TDM engine
        for (int r = 0; r < nrows; ++r) {
            if (r + 1 < nrows) {
                // DS reads of the buffer being overwritten must retire first
                // (DS vs ASYNC/TENSOR types are mutually unordered).
                asm volatile("s_wait_dscnt 0" ::: "memory");
                if ((r + 1) & 1) issue_async(r + 1, (r + 1) & 1);
                else             issue_tdm  (r + 1, (r + 1) & 1);
            }
            // Drain only the engine that filled row r; the other engine keeps
            // prefetching row r+1 in the background.
            if (r & 1) asm volatile("s_wait_asynccnt 0"  ::: "memory");
            else       asm volatile("s_wait_tensorcnt 0" ::: "memory");

            const int bufbase = (r & 1) * S;
            // Direct __shared__ indexing -> real ds_load_b32 on the scan path.
            for (int j = 0; j < S - 1; ++j) ROW_STEP(lds[bufbase + j]);
            ROW_LAST(lds[bufbase + S - 1]);
        }
    } else {
        // Generic fallback: scan straight from global memory.
        for (int r = 0; r < nrows; ++r) {
            const float* grow = x + (size_t)(tl + r) * S;
            for (int j = 0; j < S - 1; ++j) ROW_STEP(grow[j]);
            ROW_LAST(grow[S - 1]);
        }
    }
}

extern "C" void kernel_launch(void* const* d_in, const int* in_sizes, int n_in,
                              void* d_out, int out_size, void* d_ws, size_t ws_size,
                              hipStream_t stream) {
    (void)n_in; (void)out_size; (void)d_ws; (void)ws_size;
    // setup_inputs() order:
    // 0:x 1:y_obs 2:c_mean 3:c_std 4:Ini_C 5:weight_r_yom 6:weight_r_yfm
    // 7:bias_b0_yom 8:weight_b1_yom 9:epoch 10:time_lag
    const float* x      = (const float*)d_in[0];
    const float* c_mean = (const float*)d_in[2];
    const float* c_std  = (const float*)d_in[3];
    const float* ini_c  = (const float*)d_in[4];
    const float* w_yom  = (const float*)d_in[5];
    const float* w_yfm  = (const float*)d_in[6];
    const float* b0     = (const float*)d_in[7];
    const float* wb1    = (const float*)d_in[8];
    const int*   tl     = (const int*)d_in[10];

    const int B = in_sizes[1];        // y_obs is [B,1]
    const int S = in_sizes[0] / B;    // x is [B,S]

    mcpbrnn_scan_kernel<<<1, 32, 0, stream>>>(
        x, c_mean, c_std, ini_c, w_yom, w_yfm, b0, wb1, tl,
        (float*)d_out, B, S);
}